// GAT_66228395704921
// MI455X (gfx1250) — compile-verified
//
#include <hip/hip_runtime.h>
#include <cstdint>

// ---------------------------------------------------------------------------
// GAT 3-layer forward for MI455X (gfx1250, wave32).
//  - GEMMs: V_WMMA_F32_16X16X4_F32, B column-slice staged in LDS in fragment
//    order via async global->LDS loads (ASYNCcnt), A streamed with b64 loads.
//  - Edge segment-softmax: native f32 global atomics (max_num / add).
// ---------------------------------------------------------------------------

typedef __attribute__((ext_vector_type(2))) float v2f;
typedef __attribute__((ext_vector_type(8))) float v8f;

#define NEG_SLOPE 0.2f

// ---------------------------------------------------------------------------
// C[nrows x Mp] = A[nrows x K] @ B[K x Mp]   (row-major; nrows%16==0, K%4==0,
// Mp%16==0 — caller pads).  Block: 256 thr = 8 waves; blockIdx.x = col tile,
// blockIdx.y*8 + wave = row tile.  B slice (K x 16) lives in LDS, stored in
// WMMA-fragment order so each lane reads its fragment as one b64.
//
// f32 WMMA fragment layouts (ISA 7.12.2):
//   A 16x4 : lane l -> row m = l%16 ; vgpr v holds K = 2*(l/16) + v
//   B 4x16 : lane l -> col n = l%16 ; vgpr v holds K = 2*(l/16) + v
//   C 16x16: lane l -> col n = l%16 ; vgpr v holds row m = v + 8*(l/16)
// ---------------------------------------------------------------------------
__global__ void wmma_gemm_f32_lds_kernel(const float* __restrict__ A,
                                         const float* __restrict__ B,
                                         float* __restrict__ C,
                                         int nrows, int K, int Mp) {
    extern __shared__ float ldsB[];           // K*16 floats, fragment order
    const int tid  = threadIdx.x;
    const int lane = tid & 31;
    const int wib  = tid >> 5;
    const int half = lane >> 4;               // 0 or 1
    const int r    = lane & 15;
    const int col0 = blockIdx.x << 4;

    // Cooperative async fill of the B column slice into LDS.
    // (krow, col) -> lds[(krow/4)*64 + (((krow>>1)&1)*16 + col)*2 + (krow&1)]
    for (int idx = tid; idx < (K << 4); idx += blockDim.x) {
        const int krow = idx >> 4;
        const int col  = idx & 15;
        const int dst  = ((krow >> 2) << 6) + (((((krow >> 1) & 1) << 4) + col) << 1)
                         + (krow & 1);
        const uint64_t ga = (uint64_t)(const void*)(B + (size_t)krow * Mp + col0 + col);
        const unsigned la = (unsigned)(size_t)(&ldsB[dst]);
        asm volatile("global_load_async_to_lds_b32 %0, %1, off"
                     :: "v"(la), "v"(ga) : "memory");
    }
    asm volatile("s_wait_asynccnt 0x0" ::: "memory");
    __syncthreads();

    const int rt = blockIdx.y * (blockDim.x >> 5) + wib;   // row tile (uniform)
    if ((rt << 4) < nrows) {
        const float* __restrict__ Arow =
            A + (size_t)((rt << 4) + r) * K + (half << 1);
        v8f acc = {};
        for (int k = 0; k < K; k += 4) {
            __builtin_prefetch(&Arow[k + 128], 0, 1);      // A stream ahead
            const v2f a = *(const v2f*)(Arow + k);
            const v2f b = *(const v2f*)(ldsB + ((k >> 2) << 6) + (lane << 1));
            acc = __builtin_amdgcn_wmma_f32_16x16x4_f32(
                false, a, false, b, (short)0, acc, false, false);
        }
        float* __restrict__ Crow =
            C + (size_t)((rt << 4) + (half << 3)) * Mp + col0 + r;
        #pragma unroll
        for (int v = 0; v < 8; ++v)
            Crow[(size_t)v * Mp] = acc[v];
    }
}

// zero-pad W [K x M] -> Wp [K x Mp]
__global__ void pad_mat_kernel(const float* __restrict__ W, float* __restrict__ Wp,
                               int K, int M, int Mp) {
    const int gid = blockIdx.x * blockDim.x + threadIdx.x;
    if (gid >= K * Mp) return;
    const int kk = gid / Mp, c = gid % Mp;
    Wp[gid] = (c < M) ? W[kk * M + c] : 0.0f;
}

// ---------------------------------------------------------------------------
// a_s[n,h] = sum_d h[n, h*dim+d] * att_src[h,d] ; a_d likewise. h has ld cols.
// ---------------------------------------------------------------------------
__global__ void node_logits_kernel(const float* __restrict__ h, int ld,
                                   const float* __restrict__ att_src,
                                   const float* __restrict__ att_dst,
                                   float* __restrict__ aS,
                                   float* __restrict__ aD,
                                   int n, int heads, int dim) {
    const int gid = blockIdx.x * blockDim.x + threadIdx.x;
    if (gid >= n * heads) return;
    const int node = gid / heads;
    const int hh   = gid % heads;
    const float* hp = h + (size_t)node * ld + (size_t)hh * dim;
    float ss = 0.0f, dd = 0.0f;
    for (int d = 0; d < dim; ++d) {
        const float v = hp[d];
        ss += v * att_src[hh * dim + d];
        dd += v * att_dst[hh * dim + d];
    }
    aS[gid] = ss;
    aD[gid] = dd;
}

__global__ void fill_neg_inf_kernel(float* __restrict__ p, int n) {
    const int gid = blockIdx.x * blockDim.x + threadIdx.x;
    if (gid < n) p[gid] = -INFINITY;
}

__device__ __forceinline__ void edge_endpoints(const int* __restrict__ ei,
                                               int e, int E, int& s, int& d) {
    if (e < E) { s = ei[e]; d = ei[E + e]; }     // edge_index[0,:], [1,:]
    else       { s = d = e - E; }                // self-loop
}

// pass 1: m[dst,h] = max_e leaky_relu(a_s[src,h]+a_d[dst,h])
__global__ void edge_max_kernel(const int* __restrict__ ei,
                                const float* __restrict__ aS,
                                const float* __restrict__ aD,
                                float* __restrict__ m,
                                int E, int n, int heads) {
    const int gid = blockIdx.x * blockDim.x + threadIdx.x;
    if (gid >= (E + n) * heads) return;
    const int e  = gid / heads;
    const int hh = gid % heads;
    int s, d; edge_endpoints(ei, e, E, s, d);
    float v = aS[s * heads + hh] + aD[d * heads + hh];
    v = v > 0.0f ? v : NEG_SLOPE * v;
    atomicMax(&m[d * heads + hh], v);
}

// pass 2: s[dst,h] += exp(e - m[dst,h])
__global__ void edge_sum_kernel(const int* __restrict__ ei,
                                const float* __restrict__ aS,
                                const float* __restrict__ aD,
                                const float* __restrict__ m,
                                float* __restrict__ ssum,
                                int E, int n, int heads) {
    const int gid = blockIdx.x * blockDim.x + threadIdx.x;
    if (gid >= (E + n) * heads) return;
    const int e  = gid / heads;
    const int hh = gid % heads;
    int s, d; edge_endpoints(ei, e, E, s, d);
    float v = aS[s * heads + hh] + aD[d * heads + hh];
    v = v > 0.0f ? v : NEG_SLOPE * v;
    atomicAdd(&ssum[d * heads + hh], expf(v - m[d * heads + hh]));
}

// pass 3: out[dst,f] += alpha(e, f/dim) * h[src,f]  (64 threads/edge, f<HD)
__global__ void edge_agg_kernel(const int* __restrict__ ei,
                                const float* __restrict__ aS,
                                const float* __restrict__ aD,
                                const float* __restrict__ m,
                                const float* __restrict__ ssum,
                                const float* __restrict__ h, int ld,
                                float* __restrict__ out,
                                int E, int n, int heads, int dim) {
    const int HD  = heads * dim;                   // <= 64
    const int gid = blockIdx.x * blockDim.x + threadIdx.x;
    const int e   = gid >> 6;
    const int f   = gid & 63;
    if (e >= E + n || f >= HD) return;
    int s, d; edge_endpoints(ei, e, E, s, d);
    const int hh = f / dim;
    float v = aS[s * heads + hh] + aD[d * heads + hh];
    v = v > 0.0f ? v : NEG_SLOPE * v;
    const float alpha = expf(v - m[d * heads + hh]) / ssum[d * heads + hh];
    atomicAdd(&out[(size_t)d * HD + f], alpha * h[(size_t)s * ld + f]);
}

__global__ void bias_elu_kernel(float* __restrict__ x,
                                const float* __restrict__ b, int n, int HD) {
    const int gid = blockIdx.x * blockDim.x + threadIdx.x;
    if (gid >= n * HD) return;
    const float v = x[gid] + b[gid % HD];
    x[gid] = v > 0.0f ? v : (expf(v) - 1.0f);
}

// out[node,:] = log_softmax(out[node,:] + b) in-place, C <= 40
__global__ void logsoftmax_kernel(float* __restrict__ out,
                                  const float* __restrict__ b, int n, int C) {
    const int node = blockIdx.x * blockDim.x + threadIdx.x;
    if (node >= n) return;
    float vals[40];
    float mx = -INFINITY;
    for (int c = 0; c < C; ++c) {
        const float v = out[(size_t)node * C + c] + b[c];
        vals[c] = v;
        mx = v > mx ? v : mx;
    }
    float sum = 0.0f;
    for (int c = 0; c < C; ++c) sum += expf(vals[c] - mx);
    const float lse = mx + logf(sum);
    for (int c = 0; c < C; ++c) out[(size_t)node * C + c] = vals[c] - lse;
}

// ---------------------------------------------------------------------------
static inline int cdiv(long long a, long long b) { return (int)((a + b - 1) / b); }

static void run_gemm(const float* A, const float* B, float* C,
                     int nrows, int K, int Mp, hipStream_t stream) {
    dim3 grid(Mp / 16, cdiv(nrows / 16, 8));
    wmma_gemm_f32_lds_kernel<<<grid, 256, (size_t)K * 16 * sizeof(float), stream>>>(
        A, B, C, nrows, K, Mp);
}

static void run_attention(const int* ei, const float* h, int ld,
                          const float* att_src, const float* att_dst,
                          float* aS, float* aD, float* mB, float* sB,
                          float* out, int n, int E, int heads, int dim,
                          hipStream_t stream) {
    const int nh = n * heads;
    node_logits_kernel<<<cdiv(nh, 256), 256, 0, stream>>>(h, ld, att_src, att_dst,
                                                          aS, aD, n, heads, dim);
    fill_neg_inf_kernel<<<cdiv(nh, 256), 256, 0, stream>>>(mB, nh);
    hipMemsetAsync(sB, 0, (size_t)nh * sizeof(float), stream);

    const int ne = E + n;
    edge_max_kernel<<<cdiv((long long)ne * heads, 256), 256, 0, stream>>>(
        ei, aS, aD, mB, E, n, heads);
    edge_sum_kernel<<<cdiv((long long)ne * heads, 256), 256, 0, stream>>>(
        ei, aS, aD, mB, sB, E, n, heads);

    hipMemsetAsync(out, 0, (size_t)n * heads * dim * sizeof(float), stream);
    edge_agg_kernel<<<cdiv((long long)ne * 64, 256), 256, 0, stream>>>(
        ei, aS, aD, mB, sB, h, ld, out, E, n, heads, dim);
}

extern "C" void kernel_launch(void* const* d_in, const int* in_sizes, int n_in,
                              void* d_out, int out_size, void* d_ws, size_t ws_size,
                              hipStream_t stream) {
    const float* x   = (const float*)d_in[0];
    const int*   ei  = (const int*)  d_in[1];
    const float* W1  = (const float*)d_in[2];
    const float* as1 = (const float*)d_in[3];
    const float* ad1 = (const float*)d_in[4];
    const float* b1  = (const float*)d_in[5];
    const float* W2  = (const float*)d_in[6];
    const float* as2 = (const float*)d_in[7];
    const float* ad2 = (const float*)d_in[8];
    const float* b2  = (const float*)d_in[9];
    const float* W3  = (const float*)d_in[10];
    const float* as3 = (const float*)d_in[11];
    const float* ad3 = (const float*)d_in[12];
    const float* b3  = (const float*)d_in[13];

    const int FIN = 512, HID = 64, CLS = 40, CLSP = 48, H = 8, D = 8;
    const int N = in_sizes[0] / FIN;               // 100000 (multiple of 16)
    const int E = in_sizes[1] / 2;                 // 3200000

    // workspace layout (~64 MB)
    char* ws = (char*)d_ws;
    float* bufA = (float*)ws;                                       // N*64
    float* bufB = (float*)(ws + (size_t)N * HID * sizeof(float));   // N*64
    float* aS   = (float*)(ws + (size_t)N * HID * 2 * sizeof(float));
    float* aD   = aS + (size_t)N * H;
    float* mB   = aD + (size_t)N * H;
    float* sB   = mB + (size_t)N * H;
    float* W3p  = sB + (size_t)N * H;                               // 64*48

    // ---- layer 1: in=512, heads=8, dim=8, concat ----
    run_gemm(x, W1, bufA, N, FIN, HID, stream);
    run_attention(ei, bufA, HID, as1, ad1, aS, aD, mB, sB, bufB, N, E, H, D, stream);
    bias_elu_kernel<<<cdiv((long long)N * HID, 256), 256, 0, stream>>>(bufB, b1, N, HID);

    // ---- layer 2: in=64, heads=8, dim=8, concat ----
    run_gemm(bufB, W2, bufA, N, HID, HID, stream);
    run_attention(ei, bufA, HID, as2, ad2, aS, aD, mB, sB, bufB, N, E, H, D, stream);
    bias_elu_kernel<<<cdiv((long long)N * HID, 256), 256, 0, stream>>>(bufB, b2, N, HID);

    // ---- layer 3: in=64, heads=1, dim=40 (padded to 48), mean(1 head)==id ----
    pad_mat_kernel<<<cdiv(HID * CLSP, 256), 256, 0, stream>>>(W3, W3p, HID, CLS, CLSP);
    run_gemm(bufB, W3p, bufA, N, HID, CLSP, stream);
    run_attention(ei, bufA, CLSP, as3, ad3, aS, aD, mB, sB, (float*)d_out,
                  N, E, 1, CLS, stream);
    logsoftmax_kernel<<<cdiv(N, 256), 256, 0, stream>>>((float*)d_out, b3, N, CLS);
}